// Mamba2InformedElmanCell_49314814493167
// MI455X (gfx1250) — compile-verified
//
#include <hip/hip_runtime.h>

typedef __attribute__((ext_vector_type(2))) float v2f;
typedef __attribute__((ext_vector_type(4))) float v4f;
typedef __attribute__((ext_vector_type(8))) float v8f;

namespace {
constexpr int kT  = 512;
constexpr int kB  = 8;
constexpr int kH  = 16;
constexpr int kP  = 64;    // headdim
constexpr int kN  = 64;    // d_state
constexpr int kDI = 1024;  // d_inner
constexpr int kLdsStride = 68;  // padded row stride in floats (272B, 16B-aligned)
}

__global__ __launch_bounds__(32)
void mamba2_scan_wmma(const float* __restrict__ X,
                      const float* __restrict__ Bp,
                      const float* __restrict__ Cp,
                      const float* __restrict__ DT,
                      const float* __restrict__ Z,
                      const float* __restrict__ H0,
                      const float* __restrict__ DTB,
                      float* __restrict__ Out,
                      float* __restrict__ Hout)
{
  __shared__ float sH[16 * kLdsStride];

  const int l     = threadIdx.x;   // 0..31
  const int pl    = l & 15;        // row within strip
  const int half  = l >> 4;        // lane half selects n-pair offset
  const int off   = half * 2;
  const int strip = blockIdx.x;    // 0..3
  const int h     = blockIdx.y;    // 0..15
  const int b     = blockIdx.z;    // 0..7
  const int pbase = strip * 16;
  const int p     = pbase + pl;

  const float bias = DTB[h];

  // H strip in WMMA B-operand pair layout:
  //   Hc[c] = { H[p, 4c+off], H[p, 4c+off+1] },  off = 2*(lane>=16)
  v2f Hc[16];
  {
    const float* h0 = H0 + (((size_t)b * kH + h) * kP + p) * kN;
#pragma unroll
    for (int c = 0; c < 16; ++c)
      Hc[c] = *(const v2f*)(h0 + 4 * c + off);
  }

  // Stream the current H strip to Hout[tp1] via LDS transpose -> coalesced NT b128 stores.
  auto storeH = [&](int tp1) {
#pragma unroll
    for (int c = 0; c < 16; ++c)
      *(v2f*)(&sH[pl * kLdsStride + 4 * c + off]) = Hc[c];
    __syncthreads();
    float* dst = Hout + ((((size_t)tp1 * kB + b) * kH + h) * kP + pbase) * kN;
#pragma unroll
    for (int j = 0; j < 8; ++j) {
      const int row = 2 * j + half;
      const int col = pl * 4;
      v4f v = *(const v4f*)(&sH[row * kLdsStride + col]);
      __builtin_nontemporal_store(v, (v4f*)(dst + j * 128 + l * 4));
    }
    __syncthreads();
  };

  auto loadStep = [&](int t, v2f (&Bb)[16], v2f (&Cb)[16], float& xb, float& db) {
    const float* bp = Bp + ((size_t)t * kB + b) * kN;
    const float* cp = Cp + ((size_t)t * kB + b) * kN;
#pragma unroll
    for (int c = 0; c < 16; ++c) {
      Bb[c] = *(const v2f*)(bp + 4 * c + off);  // B pair in same layout as H pairs
      Cb[c] = *(const v2f*)(cp + 4 * c + off);  // C pair == replicated A-operand
    }
    xb = X[((size_t)t * kB + b) * kDI + h * kP + p];
    db = DT[((size_t)t * kB + b) * kH + h];
  };

  auto compute = [&](int t, const v2f (&Bb)[16], const v2f (&Cb)[16],
                     float xb, float db) {
    // per-head scalar decay
    const float d = 1.0f / (1.0f + __expf(-(db + bias)));
    // H = d*H + x (outer) B  -- elementwise in the pair layout
#pragma unroll
    for (int c = 0; c < 16; ++c) {
      Hc[c].x = d * Hc[c].x + xb * Bb[c].x;
      Hc[c].y = d * Hc[c].y + xb * Bb[c].y;
    }
    // y[p] = sum_n H[p,n]*C[n] via WMMA f32 16x16x4:
    //   A[m,k] = C[4c+k] (replicated rows), B[k,n] = H[pbase+n, 4c+k]
    //   => D[m,n] = y[pbase+n] in every row; lane l holds y[pbase + l%16] in D[0].
    v8f acc0 = {0.f, 0.f, 0.f, 0.f, 0.f, 0.f, 0.f, 0.f};
    v8f acc1 = {0.f, 0.f, 0.f, 0.f, 0.f, 0.f, 0.f, 0.f};
#pragma unroll
    for (int c = 0; c < 16; c += 2) {
      acc0 = __builtin_amdgcn_wmma_f32_16x16x4_f32(
          false, Cb[c],     false, Hc[c],     (short)0, acc0, false, false);
      acc1 = __builtin_amdgcn_wmma_f32_16x16x4_f32(
          false, Cb[c + 1], false, Hc[c + 1], (short)0, acc1, false, false);
    }
    const float y = acc0[0] + acc1[0];

    // gated output: y * silu(z + y); lanes 0..15 each own one p
    if (half == 0) {
      const size_t oi = ((size_t)t * kB + b) * kDI + h * kP + p;
      const float g  = Z[oi] + y;
      const float sg = g / (1.0f + __expf(-g));
      __builtin_nontemporal_store(y * sg, &Out[oi]);
    }
    storeH(t + 1);
  };

  storeH(0);  // H[0] = H0

  // two-stage software pipeline: load(t+1) overlaps compute(t)
  v2f B0[16], C0[16], B1[16], C1[16];
  float x0, x1, d0, d1;
  loadStep(0, B0, C0, x0, d0);

  for (int t = 0; t < kT; t += 2) {
    loadStep(t + 1, B1, C1, x1, d1);
    compute(t, B0, C0, x0, d0);
    if (t + 2 < kT) loadStep(t + 2, B0, C0, x0, d0);
    compute(t + 1, B1, C1, x1, d1);
  }
}

extern "C" void kernel_launch(void* const* d_in, const int* in_sizes, int n_in,
                              void* d_out, int out_size, void* d_ws, size_t ws_size,
                              hipStream_t stream) {
  const float* X   = (const float*)d_in[0];  // x        [T,B,1024]
  const float* Bpp = (const float*)d_in[1];  // B_proj   [T,B,64]
  const float* Cpp = (const float*)d_in[2];  // C_proj   [T,B,64]
  const float* DT  = (const float*)d_in[3];  // dt       [T,B,16]
  const float* Z   = (const float*)d_in[4];  // z        [T,B,1024]
  const float* H0  = (const float*)d_in[5];  // H0       [B,16,64,64]
  const float* DTB = (const float*)d_in[6];  // dt_bias  [16]

  float* Out  = (float*)d_out;                       // [T,B,1024]
  float* Hout = Out + (size_t)kT * kB * kDI;         // [(T+1),B,16,64,64]

  dim3 grid(kP / 16, kH, kB);  // strip, head, batch -> 512 single-wave blocks
  dim3 block(32);
  hipLaunchKernelGGL(mamba2_scan_wmma, grid, block, 0, stream,
                     X, Bpp, Cpp, DT, Z, H0, DTB, Out, Hout);
}